// StarINNWithILWT_72189810311206
// MI455X (gfx1250) — compile-verified
//
#include <hip/hip_runtime.h>
#include <cstddef>
#include <cstdint>

// ---------------------------------------------------------------------------
// Fused 2D CDF-5/3 lifting wavelet (reflect-1 padding), fp32, MI455X/gfx1250.
// 8x3 images of 1024x1024 -> [LL,LH,HL,HH] each 512x512, channel-concatenated.
// Memory-bound (~0.7 flop/byte, ~230 MB moved -> ~10 us at 23.3 TB/s):
// single fused pass, LDS tiling with 4-px halo, async global->LDS copies
// (ASYNCcnt path; b128 on interior tiles, clamped b32 gather on edges),
// b128 subband stores. No WMMA by design: nothing for the matrix pipes at
// this arithmetic intensity, and the output must stay fp32.
// ---------------------------------------------------------------------------

#define H   1024
#define W   1024
#define N2  512            // per-subband extent
#define TI  32             // output tile rows  (per subband)
#define TJ  32             // output tile cols  (per subband)
#define XR  (2*TI + 8)     // 72 input tile rows  (4-halo each side)
#define XC  (2*TJ + 8)     // 72 input tile cols
#define XP  XC             // 72 pitch: rows 16B-aligned for b128 async copies
#define DC  (TJ + 2)       // 34 d_row columns: global j in [j0-1, j0+TJ]
#define DP  36             // pitch (multiple of 4 floats)
#define SP  36             // pitch for s_row (TJ columns)

typedef int v4i __attribute__((vector_size(4 * sizeof(int))));
typedef __attribute__((address_space(1))) int* as1_int_ptr;
typedef __attribute__((address_space(3))) int* as3_int_ptr;
typedef __attribute__((address_space(1))) v4i* as1_v4i_ptr;
typedef __attribute__((address_space(3))) v4i* as3_v4i_ptr;

__device__ __forceinline__ int iclampi(int v, int lo, int hi) {
    v = (v < lo) ? lo : v;
    return (v > hi) ? hi : v;
}

// Reference's _rpad boundary rule: index -1 -> 1, index N2 -> N2-2.
__device__ __forceinline__ int reflectN2(int k) {
    k = (k < 0) ? -k : k;
    k = (k >= N2) ? (2 * N2 - 2 - k) : k;
    return k;
}

// Column-direction predict step on an LDS-resident row-transform plane:
// D(k) = plane[2k+1] - 0.5*(plane[2*refl(k-1)] + plane[2*refl(k+1)])
__device__ __forceinline__ float liftD(const float* base, int pitch, int col,
                                       int k, int rbase) {
    const int kl = reflectN2(k - 1);
    const int kr = reflectN2(k + 1);
    const float odd = base[(2 * k + 1 - rbase) * pitch + col];
    const float eL  = base[(2 * kl     - rbase) * pitch + col];
    const float eR  = base[(2 * kr     - rbase) * pitch + col];
    return odd - 0.5f * (eL + eR);
}

__global__ __launch_bounds__(256)
void lwt53_fused(const float* __restrict__ x, float* __restrict__ out) {
    __shared__ float xt[XR * XP];   // raw input tile           (20.25 KB)
    __shared__ float dt[XR * DP];   // row-lifting detail d_row (10.1 KB)
    __shared__ float st[XR * SP];   // row-lifting smooth s_row (10.1 KB)

    const int j0    = blockIdx.x * TJ;
    const int i0    = blockIdx.y * TI;
    const int img   = blockIdx.z;          // b*3 + c
    const int rbase = 2 * i0 - 4;          // global row of tile row 0
    const int cbase = 2 * j0 - 4;          // global col of tile col 0
    const float* __restrict__ xim = x + (size_t)img * ((size_t)H * W);
    const int t = threadIdx.x;

    // ---- Stage 0: gather the 72x72 input tile into LDS --------------------
#if defined(__HIP_DEVICE_COMPILE__) && \
    __has_builtin(__builtin_amdgcn_global_load_async_to_lds_b32) && \
    __has_builtin(__builtin_amdgcn_global_load_async_to_lds_b128)
    const bool interior = (rbase >= 0) & (rbase + XR <= H) &
                          (cbase >= 0) & (cbase + XC <= W);
    if (interior) {
        // 16B-aligned rows in both global and LDS: wide async copies.
        for (int idx = t; idx < XR * (XC / 4); idx += 256) {
            const int rr = idx / (XC / 4), q4 = (idx - rr * (XC / 4)) * 4;
            const float* gp = xim + (size_t)(rbase + rr) * W + (cbase + q4);
            __builtin_amdgcn_global_load_async_to_lds_b128(
                (as1_v4i_ptr)(gp), (as3_v4i_ptr)(&xt[rr * XP + q4]),
                /*offset=*/0, /*cpol=*/0);
        }
    } else {
        // Edge tiles: per-lane clamped gather.
        for (int idx = t; idx < XR * XC; idx += 256) {
            const int rr = idx / XC, cc = idx - rr * XC;
            const int rg = iclampi(rbase + rr, 0, H - 1);
            const int cg = iclampi(cbase + cc, 0, W - 1);
            const float* gp = xim + (size_t)rg * W + cg;
            __builtin_amdgcn_global_load_async_to_lds_b32(
                (as1_int_ptr)(gp), (as3_int_ptr)(&xt[rr * XP + cc]),
                /*offset=*/0, /*cpol=*/0);
        }
    }
#if __has_builtin(__builtin_amdgcn_s_wait_asynccnt)
    __builtin_amdgcn_s_wait_asynccnt(0);
#else
    asm volatile("s_wait_asynccnt 0" ::: "memory");
#endif
#else
    for (int idx = t; idx < XR * XC; idx += 256) {
        const int rr = idx / XC, cc = idx - rr * XC;
        const int rg = iclampi(rbase + rr, 0, H - 1);
        const int cg = iclampi(cbase + cc, 0, W - 1);
        xt[rr * XP + cc] = xim[(size_t)rg * W + cg];
    }
#endif
    __syncthreads();

    // ---- Stage A1: row predict  d_row[r, j] for j in [j0-1, j0+TJ] --------
    for (int idx = t; idx < XR * DC; idx += 256) {
        const int rr = idx / DC, jj = idx - rr * DC;
        const int j  = iclampi(j0 - 1 + jj, 0, N2 - 1);   // edge slots unread
        const float* row = &xt[rr * XP];
        const float odd = row[(2 * j + 1)            - cbase];
        const float eL  = row[2 * reflectN2(j - 1)   - cbase];
        const float eR  = row[2 * reflectN2(j + 1)   - cbase];
        dt[rr * DP + jj] = odd - 0.5f * (eL + eR);
    }
    __syncthreads();

    // ---- Stage A2: row update  s_row[r, j] for j in [j0, j0+TJ) -----------
    for (int idx = t; idx < XR * TJ; idx += 256) {
        const int rr = idx / TJ, jj = idx - rr * TJ;
        const int j  = j0 + jj;
        const int dL = reflectN2(j - 1) - (j0 - 1);       // dt local column
        const int dR = reflectN2(j + 1) - (j0 - 1);
        st[rr * SP + jj] = xt[rr * XP + (2 * j - cbase)]
                         + 0.25f * (dt[rr * DP + dL] + dt[rr * DP + dR]);
    }
    __syncthreads();

    // ---- Stage B: column lifting -> LL,LH from s_row ; HL,HH from d_row ---
    // One unit of work per thread: 4 consecutive output columns, b128 stores.
    const int b = img / 3, c = img - 3 * b;
    const size_t gstep = (size_t)3 * N2 * N2;             // subband-group step
    float* __restrict__ obase = out + (size_t)(b * 12 + c) * ((size_t)N2 * N2);

    const int ii = t >> 3;            // 0..31
    const int j4 = (t & 7) * 4;       // 0,4,...,28
    const int i  = i0 + ii;
    const int im = reflectN2(i - 1), ip = reflectN2(i + 1);

    float4 vLL, vLH, vHL, vHH;
    float* pLL = &vLL.x; float* pLH = &vLH.x;
    float* pHL = &vHL.x; float* pHH = &vHH.x;

#pragma unroll
    for (int q = 0; q < 4; ++q) {
        const int jj   = j4 + q;
        const int colS = jj;          // st column for global j0+jj
        const int colD = jj + 1;      // dt column for global j0+jj

        const float LH  = liftD(st, SP, colS, i,  rbase);
        const float LHa = liftD(st, SP, colS, im, rbase);
        const float LHb = liftD(st, SP, colS, ip, rbase);
        pLH[q] = LH;
        pLL[q] = st[(2 * i - rbase) * SP + colS] + 0.25f * (LHa + LHb);

        const float HH  = liftD(dt, DP, colD, i,  rbase);
        const float HHa = liftD(dt, DP, colD, im, rbase);
        const float HHb = liftD(dt, DP, colD, ip, rbase);
        pHH[q] = HH;
        pHL[q] = dt[(2 * i - rbase) * DP + colD] + 0.25f * (HHa + HHb);
    }

    const size_t o = (size_t)i * N2 + (j0 + j4);          // 16B-aligned
    *(float4*)(obase + o            ) = vLL;
    *(float4*)(obase + o +     gstep) = vLH;
    *(float4*)(obase + o + 2 * gstep) = vHL;
    *(float4*)(obase + o + 3 * gstep) = vHH;
}

extern "C" void kernel_launch(void* const* d_in, const int* in_sizes, int n_in,
                              void* d_out, int out_size, void* d_ws, size_t ws_size,
                              hipStream_t stream) {
    (void)in_sizes; (void)n_in; (void)out_size; (void)d_ws; (void)ws_size;
    const float* x = (const float*)d_in[0];
    float* out = (float*)d_out;
    dim3 grid(N2 / TJ, N2 / TI, 8 * 3);   // 16 x 16 x 24 = 6144 workgroups
    lwt53_fused<<<grid, 256, 0, stream>>>(x, out);
}